// KGELSA_7318624272811
// MI455X (gfx1250) — compile-verified
//
#include <hip/hip_runtime.h>
#include <math.h>

typedef __attribute__((ext_vector_type(2))) float v2f;
typedef __attribute__((ext_vector_type(8))) float v8f;

#define CDIM 64
#define PQS  128   // PQ row stride: cols 0..49 = P, 64..113 = Q

// ---------- ordered-uint float max trick (monotone bijection) ----------
__device__ __forceinline__ unsigned f2ord(float f) {
  unsigned u = __float_as_uint(f);
  return (u & 0x80000000u) ? ~u : (u | 0x80000000u);
}
__device__ __forceinline__ float ord2f(unsigned u) {
  return (u & 0x80000000u) ? __uint_as_float(u & 0x7fffffffu) : __uint_as_float(~u);
}

// ---------- prep: G = rel @ fc_w folded into B-matrix [64 x 128], cbias ----
// Bm[c][j]: j<64  -> r=j   : G[r][c]      (P columns)
//           j>=64 -> r=j-64: G[r][64+c]   (Q columns)   r>=50 -> 0
__global__ void rgat_prep(const float* __restrict__ rel,
                          const float* __restrict__ fcw,
                          const float* __restrict__ fcb,
                          float* __restrict__ Bm,
                          float* __restrict__ cbias, int R) {
  int j = threadIdx.x;               // 0..127
  int r = j & 63;
  bool valid = (r < R);
  if (j < 64) {
    float cb = 0.f;
    if (valid)
      for (int jj = 0; jj < CDIM; ++jj) cb += rel[r * CDIM + jj] * fcb[jj];
    cbias[j] = cb;
  }
  int kbase = (j < 64) ? 0 : 64;
  for (int c = 0; c < CDIM; ++c) {
    float v = 0.f;
    if (valid) {
      int k = kbase + c;
      for (int jj = 0; jj < CDIM; ++jj)
        v += rel[r * CDIM + jj] * fcw[jj * (2 * CDIM) + k];
    }
    Bm[c * PQS + j] = v;
  }
}

// ---------- WMMA GEMM: PQ[N,128] = emb[N,64] @ Bm[64,128] (f32 exact) ------
// One wave32 per 16x16 output tile. grid = (N/16, 128/16).
// A frag (16x4 f32, ISA layout): lanes 0-15 hold K=k,k+1 ; lanes 16-31 K=k+2,k+3.
// B frag mirrored: v.x = Bm[k+ka][col], v.y = Bm[k+ka+1][col].
__global__ void rgat_gemm_pq(const float* __restrict__ emb,
                             const float* __restrict__ Bm,
                             float* __restrict__ PQ, int N) {
  int tile_n = blockIdx.x * 16;
  int tile_j = blockIdx.y * 16;
  int lane   = threadIdx.x;          // 0..31
  int half   = lane >> 4;            // 0 or 1
  int lanem  = lane & 15;
  int row    = tile_n + lanem;
  if (row >= N) row = N - 1;         // clamp (N%16==0 in this problem)
  const float* arow = emb + (size_t)row * CDIM;
  int ka = half ? 2 : 0;
  int col = tile_j + lanem;

  v8f acc = {};
  #pragma unroll
  for (int k = 0; k < CDIM; k += 4) {
    v2f a, b;
    a.x = arow[k + ka];
    a.y = arow[k + ka + 1];
    b.x = Bm[(k + ka)     * PQS + col];
    b.y = Bm[(k + ka + 1) * PQS + col];
    acc = __builtin_amdgcn_wmma_f32_16x16x4_f32(
        /*neg_a=*/false, a, /*neg_b=*/false, b,
        /*c_mod=*/(short)0, acc, /*reuse_a=*/false, /*reuse_b=*/false);
  }
  // D layout: VGPR i -> row (tile_n + i + half*8), col = tile_j + lanem
  #pragma unroll
  for (int i = 0; i < 8; ++i) {
    int orow = tile_n + i + half * 8;
    if (orow < N) PQ[(size_t)orow * PQS + col] = acc[i];
  }
}

// ---------- init m (ordered -inf == 0) and denom ----------
__global__ void rgat_init_md(unsigned* __restrict__ mmax,
                             float* __restrict__ denom, int N) {
  int n = blockIdx.x * blockDim.x + threadIdx.x;
  if (n < N) { mmax[n] = 0u; denom[n] = 0.f; }
}

__global__ void rgat_zero_f(float* __restrict__ p, long long n) {
  long long i = (long long)blockIdx.x * blockDim.x + threadIdx.x;
  if (i < n) p[i] = 0.f;
}

// ---------- pass 1: logits + segment max ----------
__global__ void rgat_edge_logits(const int* __restrict__ head,
                                 const int* __restrict__ tail,
                                 const int* __restrict__ etype,
                                 const float* __restrict__ PQ,
                                 const float* __restrict__ cbias,
                                 float* __restrict__ elogit,
                                 unsigned* __restrict__ mmax, int E) {
  int e = blockIdx.x * blockDim.x + threadIdx.x;
  if (e >= E) return;
  int h = head[e], t = tail[e], r = etype[e];
  float v = PQ[(size_t)h * PQS + r] + PQ[(size_t)t * PQS + 64 + r] + cbias[r];
  v = (v > 0.f) ? v : 0.2f * v;                 // leakyrelu(0.2)
  elogit[e] = v;
  atomicMax(&mmax[h], f2ord(v));
}

// ---------- pass 2: exp + segment sum ----------
__global__ void rgat_edge_exp(const int* __restrict__ head,
                              float* __restrict__ elogit,
                              const unsigned* __restrict__ mmax,
                              float* __restrict__ denom, int E) {
  int e = blockIdx.x * blockDim.x + threadIdx.x;
  if (e >= E) return;
  int h = head[e];
  float ex = __expf(elogit[e] - ord2f(mmax[h]));
  elogit[e] = ex;
  atomicAdd(&denom[h], ex);
}

// ---------- pass 3: weighted scatter-sum (16 lanes per edge, float4) ------
__global__ void rgat_edge_scatter(const int* __restrict__ head,
                                  const int* __restrict__ tail,
                                  const float* __restrict__ elogit,
                                  const float* __restrict__ denom,
                                  const float* __restrict__ emb,
                                  float* __restrict__ agg, int E) {
  long long gid = (long long)blockIdx.x * blockDim.x + threadIdx.x;
  int e = (int)(gid >> 4);
  int q = (int)(gid & 15);
  if (e >= E) return;
  int h = head[e], t = tail[e];
  float w = elogit[e] / denom[h];
  float4 v = ((const float4*)(emb + (size_t)t * CDIM))[q];
  float* dst = agg + (size_t)h * CDIM + q * 4;
  atomicAdd(dst + 0, v.x * w);
  atomicAdd(dst + 1, v.y * w);
  atomicAdd(dst + 2, v.z * w);
  atomicAdd(dst + 3, v.w * w);
}

// ---------- finalize: residual + L2 normalize (one wave32 per node) -------
__global__ void rgat_finalize(const float* __restrict__ agg,
                              const float* __restrict__ embin,
                              float* __restrict__ embout, int N) {
  int n = blockIdx.x * (blockDim.x >> 5) + (threadIdx.x >> 5);
  int lane = threadIdx.x & 31;
  if (n >= N) return;
  size_t base = (size_t)n * CDIM;
  float a0 = agg[base + lane]      + embin[base + lane];
  float a1 = agg[base + 32 + lane] + embin[base + 32 + lane];
  float s = a0 * a0 + a1 * a1;
  #pragma unroll
  for (int off = 16; off > 0; off >>= 1) s += __shfl_xor(s, off, 32);
  float inv = 1.f / fmaxf(sqrtf(s), 1e-12f);
  embout[base + lane]      = a0 * inv;
  embout[base + 32 + lane] = a1 * inv;
}

// ---------------------------------------------------------------------------
static inline size_t alignup(size_t x) { return (x + 255) & ~(size_t)255; }

extern "C" void kernel_launch(void* const* d_in, const int* in_sizes, int n_in,
                              void* d_out, int out_size, void* d_ws, size_t ws_size,
                              hipStream_t stream) {
  const float* entity = (const float*)d_in[0];   // [N,64]
  const float* rel    = (const float*)d_in[1];   // [R,64]
  const float* fcw    = (const float*)d_in[2];   // [64,128]
  const float* fcb    = (const float*)d_in[3];   // [64]
  const int*   eidx   = (const int*)d_in[4];     // [2,E]
  const int*   etype  = (const int*)d_in[5];     // [E]

  int N = in_sizes[0] / CDIM;
  int R = in_sizes[1] / CDIM;
  int E = in_sizes[5];
  const int* head = eidx;
  const int* tail = eidx + E;

  // workspace carve (agg aliases PQ: PQ is dead after pass 1)
  char* w = (char*)d_ws;
  size_t off = 0;
  float* Bm     = (float*)(w + off); off = alignup(off + 64 * PQS * sizeof(float));
  float* cbias  = (float*)(w + off); off = alignup(off + 64 * sizeof(float));
  float* PQ     = (float*)(w + off); off = alignup(off + (size_t)N * PQS * sizeof(float));
  float* agg    = PQ;                // alias: zeroed AFTER pass 1 consumes PQ
  float* elogit = (float*)(w + off); off = alignup(off + (size_t)E * sizeof(float));
  unsigned* mmax= (unsigned*)(w + off); off = alignup(off + (size_t)N * sizeof(unsigned));
  float* denom  = (float*)(w + off); off = alignup(off + (size_t)N * sizeof(float));
  float* emb1   = (float*)(w + off); off = alignup(off + (size_t)N * CDIM * sizeof(float));
  float* outp   = (float*)d_out;

  rgat_prep<<<1, 128, 0, stream>>>(rel, fcw, fcb, Bm, cbias, R);

  const float* ein = entity;
  for (int hop = 0; hop < 2; ++hop) {
    float* eout = (hop == 0) ? emb1 : outp;

    rgat_init_md<<<(N + 255) / 256, 256, 0, stream>>>(mmax, denom, N);

    rgat_gemm_pq<<<dim3((N + 15) / 16, PQS / 16), 32, 0, stream>>>(ein, Bm, PQ, N);

    rgat_edge_logits<<<(E + 255) / 256, 256, 0, stream>>>(
        head, tail, etype, PQ, cbias, elogit, mmax, E);

    long long aggN = (long long)N * CDIM;
    rgat_zero_f<<<(unsigned)((aggN + 255) / 256), 256, 0, stream>>>(agg, aggN);

    rgat_edge_exp<<<(E + 255) / 256, 256, 0, stream>>>(head, elogit, mmax, denom, E);

    long long sc = (long long)E * 16;
    rgat_edge_scatter<<<(unsigned)((sc + 255) / 256), 256, 0, stream>>>(
        head, tail, elogit, denom, ein, agg, E);

    rgat_finalize<<<(N * 32 + 255) / 256, 256, 0, stream>>>(agg, ein, eout, N);

    ein = eout;
  }
}